// GraphConvSingle_2714419331078
// MI455X (gfx1250) — compile-verified
//
#include <hip/hip_runtime.h>
#include <hip/hip_bf16.h>

// ---------------- problem constants (match reference) ----------------
#define B_DIM   8
#define C_INF   3
#define N_DIM   32768
#define C_OUT   128
#define E_NUM   4194304
#define TOTAL   (B_DIM * N_DIM)          // 262144 nodes
#define BN_EPS  1e-5f

typedef float v2f __attribute__((ext_vector_type(2)));
typedef float v8f __attribute__((ext_vector_type(8)));

// Only check for the builtin in the DEVICE pass; amdgcn builtins are invisible
// to the host pass and tripped the previous #error spuriously.
#if defined(__HIP_DEVICE_COMPILE__) && defined(__gfx1250__)
#if !__has_builtin(__builtin_amdgcn_wmma_f32_16x16x4_f32)
#error "device pass lacks __builtin_amdgcn_wmma_f32_16x16x4_f32"
#endif
#endif

// ---------------- workspace layout (float offsets) ----------------
// [ T (TOTAL x 12) | deg (TOTAL) | stats (12 colsum + 144 Gram) | scale (128) | shift (128) | nw (E) ]
#define OFF_T       ((size_t)0)
#define SZ_T        ((size_t)TOTAL * 12)
#define OFF_DEG     (OFF_T + SZ_T)
#define OFF_STATS   (OFF_DEG + (size_t)TOTAL)
#define OFF_SCALE   (OFF_STATS + 156)
#define OFF_SHIFT   (OFF_SCALE + 128)
#define OFF_NW      (OFF_SHIFT + 128)
#define ZERO_N      (OFF_STATS + 156)    // zero T + deg + stats each launch

// ---------------- kernels ----------------

__global__ void k_zero(float* __restrict__ p, size_t n) {
    size_t i = (size_t)blockIdx.x * blockDim.x + threadIdx.x;
    if (i < n) p[i] = 0.0f;
}

// T[:,0:3] = x transposed to (B*N, C_IN)
__global__ void k_fill_tx0(const float* __restrict__ x, float* __restrict__ T) {
    int r = blockIdx.x * blockDim.x + threadIdx.x;
    if (r >= TOTAL) return;
    int b = r >> 15;             // r / N_DIM
    int n = r & (N_DIM - 1);
#pragma unroll
    for (int f = 0; f < C_INF; ++f)
        T[(size_t)r * 12 + f] = x[((size_t)b * C_INF + f) * N_DIM + n];
}

// deg[s] += w[e]  (self loops removed)
__global__ void k_degree(const int* __restrict__ ei, const float* __restrict__ w,
                         float* __restrict__ deg) {
    int e = blockIdx.x * blockDim.x + threadIdx.x;
    if (e >= E_NUM) return;
    int s = ei[e];
    int d = ei[E_NUM + e];
    if (s != d) atomicAdd(&deg[s], w[e]);
}

// deg -> deg^{-1/2} in place
__global__ void k_dinvsqrt(float* __restrict__ deg) {
    int i = blockIdx.x * blockDim.x + threadIdx.x;
    if (i >= TOTAL) return;
    float dg = deg[i];
    deg[i] = (dg > 0.0f) ? rsqrtf(fmaxf(dg, 1e-12f)) : 0.0f;
}

// nw[e] = -dis[s] * w[e] * dis[d]  (scaled Laplacian, lambda_max = 2)
__global__ void k_normw(const int* __restrict__ ei, const float* __restrict__ w,
                        const float* __restrict__ dis, float* __restrict__ nw) {
    int e = blockIdx.x * blockDim.x + threadIdx.x;
    if (e >= E_NUM) return;
    int s = ei[e];
    int d = ei[E_NUM + e];
    nw[e] = (s == d) ? 0.0f : -dis[s] * w[e] * dis[d];
}

// T[dst_rows, dcol:dcol+3] += nw[e] * T[src_rows, scol:scol+3]
__global__ void k_prop(const int* __restrict__ ei, const float* __restrict__ nw,
                       float* __restrict__ T, int scol, int dcol) {
    int e = blockIdx.x * blockDim.x + threadIdx.x;
    if (e >= E_NUM) return;
    int s = ei[e];
    int d = ei[E_NUM + e];
    float c = nw[e];
    const float* ts = T + (size_t)s * 12 + scol;
    float* td = T + (size_t)d * 12 + dcol;
    atomicAdd(&td[0], c * ts[0]);
    atomicAdd(&td[1], c * ts[1]);
    atomicAdd(&td[2], c * ts[2]);
}

// Chebyshev fixup: T[:,dcol] = 2*T[:,dcol] - T[:,scol]
__global__ void k_cheb_fix(float* __restrict__ T, int dcol, int scol) {
    int r = blockIdx.x * blockDim.x + threadIdx.x;
    if (r >= TOTAL) return;
    float* row = T + (size_t)r * 12;
#pragma unroll
    for (int f = 0; f < 3; ++f)
        row[dcol + f] = 2.0f * row[dcol + f] - row[scol + f];
}

// colsum[12] and Gram matrix (upper triangle) of T, LDS-reduced then global atomics
__global__ void k_stats(const float* __restrict__ T, float* __restrict__ stats) {
    __shared__ float ls[156];
    int tid = threadIdx.x;
    for (int i = tid; i < 156; i += blockDim.x) ls[i] = 0.0f;
    __syncthreads();
    int r = blockIdx.x * blockDim.x + tid;
    if (r < TOTAL) {
        float v[12];
        const float* row = T + (size_t)r * 12;
#pragma unroll
        for (int j = 0; j < 12; ++j) v[j] = row[j];
#pragma unroll
        for (int j = 0; j < 12; ++j) {
            atomicAdd(&ls[j], v[j]);                    // colsum
#pragma unroll
            for (int j2 = j; j2 < 12; ++j2)
                atomicAdd(&ls[12 + j * 12 + j2], v[j] * v[j2]);
        }
    }
    __syncthreads();
    if (tid < 156) atomicAdd(&stats[tid], ls[tid]);
}

// per-channel BN scale/shift computed analytically from colsum + Gram
__global__ void k_prep(const float* __restrict__ stats, const float* __restrict__ W,
                       const float* __restrict__ bias, const float* __restrict__ gamma,
                       const float* __restrict__ beta, float* __restrict__ scale,
                       float* __restrict__ shift) {
    int c = threadIdx.x;
    if (c >= C_OUT) return;
    float wc[12];
#pragma unroll
    for (int j = 0; j < 12; ++j) wc[j] = W[(size_t)j * C_OUT + c];
    const float* colsum = stats;
    const float* G = stats + 12;
    float cs = 0.0f;
#pragma unroll
    for (int j = 0; j < 12; ++j) cs += colsum[j] * wc[j];
    float quad = 0.0f;
#pragma unroll
    for (int j = 0; j < 12; ++j) {
        quad += wc[j] * wc[j] * G[j * 12 + j];
#pragma unroll
        for (int j2 = j + 1; j2 < 12; ++j2)
            quad += 2.0f * wc[j] * wc[j2] * G[j * 12 + j2];
    }
    const float invT = 1.0f / (float)TOTAL;
    float b  = bias[c];
    float mu = cs * invT + b;
    float eo2 = (quad + 2.0f * b * cs) * invT + b * b;
    float var = eo2 - mu * mu;
    float s = gamma[c] * rsqrtf(var + BN_EPS);
    scale[c] = s;
    shift[c] = (b - mu) * s + beta[c];
}

// Fused GEMM (T @ W via V_WMMA_F32_16X16X4_F32) + BN + ReLU + transposed store.
// One wave handles 128 consecutive nodes; LDS-staged coalesced channel-major writes.
__global__ void __launch_bounds__(128)
k_gemm_bn(const float* __restrict__ T, const float* __restrict__ W,
          const float* __restrict__ scale, const float* __restrict__ shift,
          float* __restrict__ out) {
    __shared__ float lds[4][128 * 17];   // per-wave staging, padded to kill bank conflicts
    const int lane = threadIdx.x & 31;
    const int wv   = threadIdx.x >> 5;
    const int hi   = lane >> 4;          // half-wave select
    const int ml   = lane & 15;          // row / column within tile
    const int row0 = (blockIdx.x * 4 + wv) * 128;
    const int b    = row0 >> 15;         // row0 / N_DIM
    const int n0   = row0 & (N_DIM - 1);

    // preload B operands: 8 channel-blocks x 3 k-steps, 2 VGPRs each
    // B 4x16 layout: vgpr j, lanes0-15 -> K=j, lanes16-31 -> K=j+2, N=lane&15
    v2f Breg[8][3];
#pragma unroll
    for (int cb = 0; cb < 8; ++cb)
#pragma unroll
        for (int kk = 0; kk < 3; ++kk) {
            int kx = 4 * kk + 2 * hi;
            int c  = cb * 16 + ml;
            v2f t;
            t.x = W[(size_t)kx * C_OUT + c];
            t.y = W[(size_t)(kx + 1) * C_OUT + c];
            Breg[cb][kk] = t;
        }
    float sc[8], sh[8];
#pragma unroll
    for (int cb = 0; cb < 8; ++cb) { sc[cb] = scale[cb * 16 + ml]; sh[cb] = shift[cb * 16 + ml]; }

#pragma unroll
    for (int cb = 0; cb < 8; ++cb) {
#pragma unroll
        for (int t = 0; t < 8; ++t) {
            // A 16x4 layout: lane holds row M=ml, K = {2*hi, 2*hi+1} per k-step
            int node = row0 + t * 16 + ml;
            const float* trow = T + (size_t)node * 12 + 2 * hi;
            v2f a0 = *(const v2f*)(trow + 0);
            v2f a1 = *(const v2f*)(trow + 4);
            v2f a2 = *(const v2f*)(trow + 8);
            v8f acc = {};
            acc = __builtin_amdgcn_wmma_f32_16x16x4_f32(false, a0, false, Breg[cb][0],
                                                        (short)0, acc, false, false);
            acc = __builtin_amdgcn_wmma_f32_16x16x4_f32(false, a1, false, Breg[cb][1],
                                                        (short)0, acc, false, false);
            acc = __builtin_amdgcn_wmma_f32_16x16x4_f32(false, a2, false, Breg[cb][2],
                                                        (short)0, acc, false, false);
            // C layout: vgpr r, lane -> M = r + 8*hi, N = ml. Apply BN+ReLU, stage to LDS.
#pragma unroll
            for (int r = 0; r < 8; ++r) {
                float v = fmaxf(acc[r] * sc[cb] + sh[cb], 0.0f);
                int nl = t * 16 + r + 8 * hi;
                lds[wv][nl * 17 + ml] = v;
            }
        }
        // coalesced channel-major store: out[b][cb*16+ch][n0+nl]
        // (same-wave LDS RAW; DS ops are in-order per wave, compiler inserts s_wait_dscnt)
        for (int i = lane; i < 2048; i += 32) {
            int ch = i >> 7;
            int nl = i & 127;
            out[((size_t)(b * C_OUT + cb * 16 + ch)) * N_DIM + (n0 + nl)] =
                lds[wv][nl * 17 + ch];
        }
    }
}

// ---------------- launch ----------------
extern "C" void kernel_launch(void* const* d_in, const int* in_sizes, int n_in,
                              void* d_out, int out_size, void* d_ws, size_t ws_size,
                              hipStream_t stream) {
    const float* x     = (const float*)d_in[0];
    const int*   ei    = (const int*)d_in[1];
    const float* ew    = (const float*)d_in[2];
    const float* W     = (const float*)d_in[3];   // (4,3,128) row-major == (12,128)
    const float* bias  = (const float*)d_in[4];
    const float* gamma = (const float*)d_in[5];
    const float* beta  = (const float*)d_in[6];
    float* out = (float*)d_out;
    float* ws  = (float*)d_ws;

    float* T     = ws + OFF_T;
    float* deg   = ws + OFF_DEG;
    float* stats = ws + OFF_STATS;
    float* scl   = ws + OFF_SCALE;
    float* shf   = ws + OFF_SHIFT;
    float* nw    = ws + OFF_NW;

    const int TPB = 256;
    // zero T + deg + stats (must happen every call: atomics accumulate into them)
    {
        size_t n = ZERO_N;
        int blocks = (int)((n + TPB - 1) / TPB);
        k_zero<<<blocks, TPB, 0, stream>>>(ws, n);
    }
    k_fill_tx0<<<TOTAL / TPB, TPB, 0, stream>>>(x, T);
    k_degree<<<E_NUM / TPB, TPB, 0, stream>>>(ei, ew, deg);
    k_dinvsqrt<<<TOTAL / TPB, TPB, 0, stream>>>(deg);
    k_normw<<<E_NUM / TPB, TPB, 0, stream>>>(ei, ew, deg, nw);

    // Chebyshev recursion over 12-wide basis stored in T
    k_prop<<<E_NUM / TPB, TPB, 0, stream>>>(ei, nw, T, 0, 3);            // T1 = L T0
    k_prop<<<E_NUM / TPB, TPB, 0, stream>>>(ei, nw, T, 3, 6);            // tmp = L T1
    k_cheb_fix<<<TOTAL / TPB, TPB, 0, stream>>>(T, 6, 0);                // T2 = 2 tmp - T0
    k_prop<<<E_NUM / TPB, TPB, 0, stream>>>(ei, nw, T, 6, 9);            // tmp = L T2
    k_cheb_fix<<<TOTAL / TPB, TPB, 0, stream>>>(T, 9, 3);                // T3 = 2 tmp - T1

    // analytic BatchNorm statistics (avoids materializing the 134MB activation twice)
    k_stats<<<TOTAL / TPB, TPB, 0, stream>>>(T, stats);
    k_prep<<<1, 128, 0, stream>>>(stats, W, bias, gamma, beta, scl, shf);

    // fused WMMA GEMM + BN + ReLU + transposed store
    k_gemm_bn<<<TOTAL / (128 * 4), 128, 0, stream>>>(T, W, scl, shf, out);
}